// BoundaryDiceLoss_82171314307268
// MI455X (gfx1250) — compile-verified
//
#include <hip/hip_runtime.h>
#include <hip/hip_bf16.h>
#include <stdint.h>

// Problem geometry (fixed by the reference)
#define NB   4
#define ND   128
#define V_   (128*128*128)       // 2,097,152 voxels per sample (= 1<<21)
#define WPR  4                   // 32-bit words per x-row (128/32)
#define WPS  (ND*ND*WPR)         // 65,536 packed words per sample

typedef __attribute__((ext_vector_type(2))) float v2f;
typedef __attribute__((ext_vector_type(8))) float v8f;

// ---------------------------------------------------------------------------
// Pass 0: zero the 16 per-sample accumulators (inter, sum_p, sum_t, count) x4
// ---------------------------------------------------------------------------
__global__ void init_accum(float* __restrict__ accum) {
  if (threadIdx.x < 16) accum[threadIdx.x] = 0.0f;
}

// ---------------------------------------------------------------------------
// Pass 1: softmax-prob + bit-packed binarization (wave32 ballot)
//   probs[b,s] = sigmoid(o1-o0); predM/tgtM are 1 bit per voxel, 32/word.
//   Fully coalesced; this pass carries the mandatory 100.5 MB HBM read.
// ---------------------------------------------------------------------------
__global__ void pass_mask(const float* __restrict__ logits,
                          const float* __restrict__ target,
                          float* __restrict__ probs,
                          uint32_t* __restrict__ predM,
                          uint32_t* __restrict__ tgtM) {
  const int idx = blockIdx.x * blockDim.x + threadIdx.x;   // 0 .. 4*V_-1
  const int b = idx >> 21;
  const int s = idx & (V_ - 1);
  const float o0 = logits[(size_t)(b * 2)     * V_ + s];
  const float o1 = logits[(size_t)(b * 2 + 1) * V_ + s];
  const float t  = target[(size_t)b * V_ + s];
  const float p  = 1.0f / (1.0f + __expf(o0 - o1));        // softmax class-1
  probs[idx] = p;
  // wave32-native bit packing: lane i <-> x = 32*w + i
  const uint32_t pb = __builtin_amdgcn_ballot_w32(o1 > o0);   // p > 0.5
  const uint32_t tb = __builtin_amdgcn_ballot_w32(t > 0.5f);
  if ((threadIdx.x & 31) == 0) {
    predM[idx >> 5] = pb;
    tgtM [idx >> 5] = tb;
  }
}

// ---------------------------------------------------------------------------
// Pass 2: boundary detection on packed bits (both sides of any axis change),
// matching jnp.diff + pad-before/pad-after. Union of pred/tgt boundaries
// (dilation distributes over union).
// ---------------------------------------------------------------------------
__device__ __forceinline__ uint32_t bnd_word(const uint32_t* __restrict__ M,
                                             int b, int z, int y, int wx) {
  auto W = [&](int zz, int yy, int wwx) -> uint32_t {
    return M[(size_t)((b * ND + zz) * ND + yy) * WPR + wwx];
  };
  const uint32_t m    = W(z, y, wx);
  const uint32_t prev = (wx > 0)       ? W(z, y, wx - 1) : 0u;
  const uint32_t next = (wx < WPR - 1) ? W(z, y, wx + 1) : 0u;
  const uint32_t ml = (m << 1) | (prev >> 31);   // bit i := m(x-1)
  const uint32_t mr = (m >> 1) | (next << 31);   // bit i := m(x+1)
  uint32_t d  = (m ^ ml) & ((wx == 0)       ? 0xFFFFFFFEu : 0xFFFFFFFFu);
  d          |= (m ^ mr) & ((wx == WPR - 1) ? 0x7FFFFFFFu : 0xFFFFFFFFu);
  if (y > 0)      d |= m ^ W(z, y - 1, wx);
  if (y < ND - 1) d |= m ^ W(z, y + 1, wx);
  if (z > 0)      d |= m ^ W(z - 1, y, wx);
  if (z < ND - 1) d |= m ^ W(z + 1, y, wx);
  return d;
}

__global__ void pass_boundary(const uint32_t* __restrict__ predM,
                              const uint32_t* __restrict__ tgtM,
                              uint32_t* __restrict__ bnd) {
  const int w = blockIdx.x * blockDim.x + threadIdx.x;     // 0 .. NB*WPS-1
  const int b  = w >> 16;                                  // WPS = 1<<16
  const int r  = w & (WPS - 1);
  const int wx = r & (WPR - 1);
  const int y  = (r >> 2) & (ND - 1);
  const int z  = r >> 9;
  bnd[w] = bnd_word(predM, b, z, y, wx) | bnd_word(tgtM, b, z, y, wx);
}

// ---------------------------------------------------------------------------
// Pass 3: word-level dilation by the Euclidean ball of radius 2.
//   x-extent e per (dz,dy): e=2 if dz=dy=0; e=1 if dz^2+dy^2 in {1,2,3};
//   e=0 if dz^2+dy^2==4. Cross-word carries; zero-fill at volume edges
//   reproduces the zero-padded SAME convolution exactly.
// Operates entirely on the 1MB L2-resident packed arrays: 32 voxels/thread.
// ---------------------------------------------------------------------------
__global__ void pass_dilate(const uint32_t* __restrict__ bnd,
                            uint32_t* __restrict__ region) {
  const int w = blockIdx.x * blockDim.x + threadIdx.x;     // 0 .. NB*WPS-1
  const int b  = w >> 16;
  const int r  = w & (WPS - 1);
  const int wx = r & (WPR - 1);
  const int y  = (r >> 2) & (ND - 1);
  const int z  = r >> 9;

  const uint32_t* __restrict__ bndS = bnd + (size_t)b * WPS;
  uint32_t acc = 0u;
#pragma unroll
  for (int dz = -2; dz <= 2; ++dz) {
    const int zz = z + dz;
    if ((unsigned)zz >= (unsigned)ND) continue;
#pragma unroll
    for (int dy = -2; dy <= 2; ++dy) {
      const int sq = dz * dz + dy * dy;
      if (sq > 4) continue;                                // outside ball
      const int yy = y + dy;
      if ((unsigned)yy >= (unsigned)ND) continue;
      const uint32_t* __restrict__ row = bndS + (size_t)(zz * ND + yy) * WPR;
      const uint32_t c = row[wx];
      const int e = (sq == 0) ? 2 : ((sq < 4) ? 1 : 0);
      if (e == 0) {
        acc |= c;
      } else {
        const uint32_t prev = (wx > 0)       ? row[wx - 1] : 0u;
        const uint32_t next = (wx < WPR - 1) ? row[wx + 1] : 0u;
        uint32_t v = c
                   | ((c << 1) | (prev >> 31))
                   | ((c >> 1) | (next << 31));
        if (e == 2) {
          v |= ((c << 2) | (prev >> 30))
             | ((c >> 2) | (next << 30));
        }
        acc |= v;
      }
    }
  }
  region[w] = acc;
}

// ---------------------------------------------------------------------------
// Pass 4: streaming masked-dice reduction, 4 voxels/thread via b128 loads.
// Per quad: one region nibble (word shared by 8 consecutive lanes), one float4
// prob load (L2-hit) and one float4 target load -> 36 bytes per wait instead
// of 9. Wave-level reduction via WMMA, then atomicAdd to 16 accumulators.
// ---------------------------------------------------------------------------
__global__ void pass_band_reduce(const float* __restrict__ probs,
                                 const float* __restrict__ target,
                                 const uint32_t* __restrict__ region,
                                 float* __restrict__ accum) {
  const int b = blockIdx.y;
  const uint32_t* __restrict__ rgS  = region + (size_t)b * WPS;
  const float4*   __restrict__ prS4 = (const float4*)(probs  + (size_t)b * V_);
  const float4*   __restrict__ tgS4 = (const float4*)(target + (size_t)b * V_);

  float a_inter = 0.0f, a_p = 0.0f, a_t = 0.0f, a_cnt = 0.0f;

  const int nquad  = V_ / 4;                               // 524,288
  const int stride = gridDim.x * blockDim.x;               // 16,384 (divides)
  for (int q = blockIdx.x * blockDim.x + threadIdx.x; q < nquad; q += stride) {
    __builtin_prefetch(&prS4[q + stride], 0, 1);           // global_prefetch_b8
    __builtin_prefetch(&tgS4[q + stride], 0, 1);
    const int s = q << 2;                                  // voxel index (4-al.)
    const uint32_t rw  = rgS[s >> 5];                      // shared by 8 lanes
    const uint32_t nib = (rw >> (s & 31)) & 0xFu;          // this quad's bits
    const float4 p = prS4[q];
    const float4 t = tgS4[q];
    const float m0 = (float)( nib       & 1u);
    const float m1 = (float)((nib >> 1) & 1u);
    const float m2 = (float)((nib >> 2) & 1u);
    const float m3 = (float)((nib >> 3) & 1u);
    const float pm0 = p.x * m0, pm1 = p.y * m1;
    const float pm2 = p.z * m2, pm3 = p.w * m3;
    a_inter = __fmaf_rn(pm0, t.x, __fmaf_rn(pm1, t.y,
              __fmaf_rn(pm2, t.z, __fmaf_rn(pm3, t.w, a_inter))));
    a_p    += (pm0 + pm1) + (pm2 + pm3);
    a_t     = __fmaf_rn(t.x, m0, __fmaf_rn(t.y, m1,
              __fmaf_rn(t.z, m2, __fmaf_rn(t.w, m3, a_t))));
    a_cnt  += (m0 + m1) + (m2 + m3);
  }

  // Wave-level reduction via v_wmma_f32_16x16x4_f32 with B == all-ones.
  // A layout: lanes 0-15 supply A[m,0]; lanes 16-31 supply A[m,2].
  // With B all-ones (layout-independent), D[m,n] = val(lane m)+val(lane m+16);
  // column 0 of D lives in lanes 0 (M=0..7) and 16 (M=8..15).
  // EXEC is all-ones here: full 256-thread blocks, uniform trip counts.
  v2f bones = {1.0f, 1.0f};
  v8f z8    = {0.f, 0.f, 0.f, 0.f, 0.f, 0.f, 0.f, 0.f};
  v2f a0 = {a_inter, 0.0f};
  v2f a1 = {a_p,     0.0f};
  v2f a2 = {a_t,     0.0f};
  v2f a3 = {a_cnt,   0.0f};
  v8f c0 = __builtin_amdgcn_wmma_f32_16x16x4_f32(false, a0, false, bones, (short)0, z8, false, false);
  v8f c1 = __builtin_amdgcn_wmma_f32_16x16x4_f32(false, a1, false, bones, (short)0, z8, false, false);
  v8f c2 = __builtin_amdgcn_wmma_f32_16x16x4_f32(false, a2, false, bones, (short)0, z8, false, false);
  v8f c3 = __builtin_amdgcn_wmma_f32_16x16x4_f32(false, a3, false, bones, (short)0, z8, false, false);

  const int lane = threadIdx.x & 31;
  if (lane == 0 || lane == 16) {
    float s0 = 0.f, s1 = 0.f, s2 = 0.f, s3 = 0.f;
#pragma unroll
    for (int j = 0; j < 8; ++j) { s0 += c0[j]; s1 += c1[j]; s2 += c2[j]; s3 += c3[j]; }
    atomicAdd(&accum[b * 4 + 0], s0);
    atomicAdd(&accum[b * 4 + 1], s1);
    atomicAdd(&accum[b * 4 + 2], s2);
    atomicAdd(&accum[b * 4 + 3], s3);
  }
}

// ---------------------------------------------------------------------------
// Pass 5: dice + mean -> scalar
// ---------------------------------------------------------------------------
__global__ void finalize_loss(const float* __restrict__ accum,
                              float* __restrict__ out) {
  if (threadIdx.x == 0) {
    float loss = 0.0f;
    for (int b = 0; b < NB; ++b) {
      const float inter = accum[b * 4 + 0];
      const float sp    = accum[b * 4 + 1];
      const float st    = accum[b * 4 + 2];
      const float cnt   = accum[b * 4 + 3];
      const float dice  = (2.0f * inter + 1e-5f) / (sp + st + 1e-5f);
      loss += (cnt > 0.0f) ? (1.0f - dice) : 0.0f;
    }
    out[0] = loss * (1.0f / (float)NB);
  }
}

// ---------------------------------------------------------------------------
// Launch: logits = d_in[0] (4,2,128^3 f32), target = d_in[1] (4,1,128^3 f32)
// Workspace layout (all L2-resident on MI455X's 192MB L2):
//   [0, 32MB)        probs  (f32, 4*128^3)
//   [+0MB, +1MB)     predM  (packed bits)
//   [+1MB, +2MB)     tgtM
//   [+2MB, +3MB)     bnd
//   [+3MB, +4MB)     region (packed band mask)
//   [+4MB, +4MB+64B) accum  (16 f32)
// ---------------------------------------------------------------------------
extern "C" void kernel_launch(void* const* d_in, const int* in_sizes, int n_in,
                              void* d_out, int out_size, void* d_ws, size_t ws_size,
                              hipStream_t stream) {
  const float* logits = (const float*)d_in[0];
  const float* target = (const float*)d_in[1];

  char* ws = (char*)d_ws;
  const size_t PROBS_BYTES = (size_t)NB * V_ * sizeof(float);     // 33,554,432
  const size_t MASK_BYTES  = (size_t)NB * WPS * sizeof(uint32_t); //  1,048,576
  float*    probs  = (float*)ws;
  uint32_t* predM  = (uint32_t*)(ws + PROBS_BYTES);
  uint32_t* tgtM   = (uint32_t*)(ws + PROBS_BYTES + 1 * MASK_BYTES);
  uint32_t* bnd    = (uint32_t*)(ws + PROBS_BYTES + 2 * MASK_BYTES);
  uint32_t* region = (uint32_t*)(ws + PROBS_BYTES + 3 * MASK_BYTES);
  float*    accum  = (float*)(ws + PROBS_BYTES + 4 * MASK_BYTES);

  init_accum<<<1, 32, 0, stream>>>(accum);

  // Pass 1: 8,388,608 threads (streams 100.5 MB from HBM — the roofline term)
  pass_mask<<<(NB * V_) / 256, 256, 0, stream>>>(logits, target, probs, predM, tgtM);

  // Pass 2+3: 262,144 packed words each, entirely in L2
  pass_boundary<<<(NB * WPS) / 256, 256, 0, stream>>>(predM, tgtM, bnd);
  pass_dilate  <<<(NB * WPS) / 256, 256, 0, stream>>>(bnd, region);

  // Pass 4: streaming reduce; 64 blocks x 256 threads per sample
  dim3 g4(64, NB);
  pass_band_reduce<<<g4, 256, 0, stream>>>(probs, target, region, accum);

  finalize_loss<<<1, 32, 0, stream>>>(accum, (float*)d_out);
}